// Loss_37735582663282
// MI455X (gfx1250) — compile-verified
//
#include <hip/hip_runtime.h>
#include <stdint.h>

// ---- problem constants (B=8192 samples of 64x64 f32) ----
#define NSAMP 8192
#define HW    4096   // 64*64 elements per sample
#define TPB   256    // 8 wave32s per workgroup, one workgroup per sample

typedef unsigned int v4u __attribute__((ext_vector_type(4)));
typedef int          v8i __attribute__((ext_vector_type(8)));
typedef int          v4i __attribute__((ext_vector_type(4)));

#if defined(__has_builtin)
#if __has_builtin(__builtin_amdgcn_tensor_load_to_lds)
#define USE_TDM 1
#endif
#endif
#ifndef USE_TDM
#define USE_TDM 0
#endif

// One workgroup per sample. Single pass over x:
//  - TDM DMA the 16KB sample into LDS (tensor_load_to_lds, TENSORcnt)
//  - each thread reduces 16 elements (k = t&63 constant per thread)
//  - fused argmax + moment sums -> per-sample loss in d_ws
__global__ __launch_bounds__(TPB) void heatmap_loss_kernel(
    const float* __restrict__ x, float* __restrict__ per_sample) {
  __shared__ float tile[HW];
  __shared__ float rs0[8], rsj[8], rsj2[8], rsk[8], rsk2[8], rmv[8];
  __shared__ int   rmi[8];

  const int t = threadIdx.x;
  const unsigned b = blockIdx.x;

  // Never-executed store: keeps LLVM from treating `tile` as store-free
  // (the TDM writes it behind the compiler's back via the D# descriptor).
  if (b > 0x0FFFFFFFu) tile[t] = 0.0f;

#if USE_TDM
  if (t < 32) {  // wave 0 issues the tensor DMA (EXEC all-ones in this wave)
    uint64_t ga  = (uint64_t)(uintptr_t)x + (uint64_t)b * (HW * 4u);
    uint32_t lds = (uint32_t)(uintptr_t)&tile[0];  // low 32 bits = LDS byte addr

    v4u g0;
    g0.x = 1u;                                        // count=1 (valid D#)
    g0.y = lds;                                       // lds_addr (bytes)
    g0.z = (uint32_t)ga;                              // global_addr[31:0]
    g0.w = (uint32_t)((ga >> 32) & 0x01FFFFFFu)       // global_addr[56:32]
           | 0x80000000u;                             // type=2 ("image")

    v8i g1;
    g1[0] = 0x00020000;            // data_size=2 (4 bytes), no pad/iterate/mcast
    g1[1] = (int)(4096u << 16);    // tensor_dim0[15:0]=4096 in bits[63:48]
    g1[2] = (int)(1u << 16);       // tensor_dim0 hi=0 ; tensor_dim1[15:0]=1
    g1[3] = (int)(4096u << 16);    // tensor_dim1 hi=0 ; tile_dim0=4096
    g1[4] = 1;                     // tile_dim1=1 ; tile_dim2=0
    g1[5] = 4096;                  // tensor_dim0_stride[31:0]
    g1[6] = (int)(4096u << 16);    // d0_stride hi=0 ; tensor_dim1_stride lo16
    g1[7] = 0;                     // tensor_dim1_stride hi

    v4i z4 = {0, 0, 0, 0};
#if defined(__clang_major__) && (__clang_major__ >= 23)
    v8i z8 = {0, 0, 0, 0, 0, 0, 0, 0};
    __builtin_amdgcn_tensor_load_to_lds(g0, g1, z4, z4, z8, 0);
#else
    __builtin_amdgcn_tensor_load_to_lds(g0, g1, z4, z4, 0);
#endif
    __builtin_amdgcn_s_wait_tensorcnt(0);
  }
#else
  {  // fallback: cooperative vector loads into LDS
    const float* xs = x + (size_t)b * HW;
#pragma unroll
    for (int i = 0; i < 16; ++i) tile[t + TPB * i] = xs[t + TPB * i];
  }
#endif
  __syncthreads();

  // Each thread handles elements e = t + 256*i. Since 256 % 64 == 0,
  // column k = t & 63 is constant per thread; row j = (t>>6) + 4*i.
  float maxv = -__builtin_inff();
  int   maxi = 0;
  float s0 = 0.0f, sj = 0.0f, sj2 = 0.0f;
  const float j0 = (float)(t >> 6);
#pragma unroll
  for (int i = 0; i < 16; ++i) {
    const int   e  = t + TPB * i;
    const float v  = tile[e];
    const float jf = j0 + 4.0f * (float)i;
    s0  += v;
    sj   = fmaf(jf, v, sj);
    sj2  = fmaf(jf * jf, v, sj2);
    if (v > maxv) { maxv = v; maxi = e; }  // ascending e -> keeps first max
  }
  const float kf  = (float)(t & 63);
  float sk  = kf * s0;        // Sigma k*x for this thread's k-constant slice
  float sk2 = kf * kf * s0;

  // wave32 butterfly reduction (5 sums + lexicographic argmax)
#pragma unroll
  for (int m = 16; m >= 1; m >>= 1) {
    s0  += __shfl_xor(s0,  m, 32);
    sj  += __shfl_xor(sj,  m, 32);
    sj2 += __shfl_xor(sj2, m, 32);
    sk  += __shfl_xor(sk,  m, 32);
    sk2 += __shfl_xor(sk2, m, 32);
    const float ov = __shfl_xor(maxv, m, 32);
    const int   oi = __shfl_xor(maxi, m, 32);
    if (ov > maxv || (ov == maxv && oi < maxi)) { maxv = ov; maxi = oi; }
  }
  const int wave = t >> 5;
  if ((t & 31) == 0) {
    rs0[wave] = s0;  rsj[wave] = sj;  rsj2[wave] = sj2;
    rsk[wave] = sk;  rsk2[wave] = sk2;
    rmv[wave] = maxv; rmi[wave] = maxi;
  }
  __syncthreads();

  if (t == 0) {
    float S0 = rs0[0], SJ = rsj[0], SJ2 = rsj2[0], SK = rsk[0], SK2 = rsk2[0];
    float MV = rmv[0];
    int   MI = rmi[0];
#pragma unroll
    for (int w = 1; w < 8; ++w) {
      S0 += rs0[w]; SJ += rsj[w]; SJ2 += rsj2[w]; SK += rsk[w]; SK2 += rsk2[w];
      if (rmv[w] > MV || (rmv[w] == MV && rmi[w] < MI)) { MV = rmv[w]; MI = rmi[w]; }
    }
    const float mx = (float)(MI >> 6);   // flat // 64  (row)
    const float my = (float)(MI & 63);   // flat %  64  (col)
    per_sample[b] = (mx * mx + my * my) * S0
                    - 2.0f * mx * SJ - 2.0f * my * SK + SJ2 + SK2;
  }
}

// Deterministic 8192 -> 1 reduction (double accumulation, fixed tree).
__global__ __launch_bounds__(TPB) void final_reduce_kernel(
    const float* __restrict__ per_sample, float* __restrict__ out) {
  __shared__ double red[8];
  const int t = threadIdx.x;
  double acc = 0.0;
#pragma unroll
  for (int i = 0; i < NSAMP / TPB; ++i) acc += (double)per_sample[t + TPB * i];
#pragma unroll
  for (int m = 16; m >= 1; m >>= 1) acc += __shfl_xor(acc, m, 32);
  if ((t & 31) == 0) red[t >> 5] = acc;
  __syncthreads();
  if (t == 0) {
    double s = 0.0;
#pragma unroll
    for (int w = 0; w < 8; ++w) s += red[w];
    out[0] = (float)s;
  }
}

extern "C" void kernel_launch(void* const* d_in, const int* in_sizes, int n_in,
                              void* d_out, int out_size, void* d_ws, size_t ws_size,
                              hipStream_t stream) {
  const float* x   = (const float*)d_in[0];
  float*       ws  = (float*)d_ws;   // NSAMP per-sample losses (32 KB)
  float*       out = (float*)d_out;  // 1 float
  (void)in_sizes; (void)n_in; (void)out_size; (void)ws_size;

  heatmap_loss_kernel<<<NSAMP, TPB, 0, stream>>>(x, ws);
  final_reduce_kernel<<<1, TPB, 0, stream>>>(ws, out);
}